// GraphAttentionLayer_16956530885144
// MI455X (gfx1250) — compile-verified
//
#include <hip/hip_runtime.h>

typedef __attribute__((ext_vector_type(2))) float v2f;
typedef __attribute__((ext_vector_type(8))) float v8f;

#define GAT_ALPHA 0.2f
#define GAT_NEG_BIG -9e15f

constexpr int Bn = 32;                 // batch
constexpr int Nn = 512;                // nodes
constexpr int Tn = 12;                 // time steps
constexpr int Fn = 64;                 // features (in == out)
constexpr int Rn = Bn * Tn * Nn;       // 196608 rows in (b,t,n) order
constexpr int LDSW = 80;               // padded LDS row stride (floats): conflict-free

__device__ __forceinline__ v8f wmma_f32x4(v2f a, v2f b, v8f c) {
  // D = A(16x4 f32) * B(4x16 f32) + C(16x16 f32), exact fp32 matrix pipe
  return __builtin_amdgcn_wmma_f32_16x16x4_f32(false, a, false, b, (short)0, c,
                                               false, false);
}

// Raw LDS byte offset (wave-relative) for async-to-LDS VDST operand
__device__ __forceinline__ unsigned int lds_off(const void* p) {
  return (unsigned int)(unsigned long long)(
      __attribute__((address_space(3))) const void*)p;
}

// Issue one 64x64-float chunk (16 KB) as 8 async b128 loads per thread.
// 128 threads * 8 = 1024 float4 = whole chunk; tracked by ASYNCcnt.
__device__ __forceinline__ void issue_chunk_async(const float* __restrict__ src,
                                                  float* dstBuf, int tid) {
#pragma unroll
  for (int j = 0; j < 8; ++j) {
    int idx = tid + j * 128;                    // float4 index 0..1023
    int row = idx >> 4, q = idx & 15;
    unsigned long long g =
        (unsigned long long)(const void*)(src + (size_t)row * Fn + q * 4);
    unsigned int l = lds_off(dstBuf + row * LDSW + q * 4);
    asm volatile("global_load_async_to_lds_b128 %0, %1, off"
                 :
                 : "v"(l), "v"(g)
                 : "memory");
  }
}

// ---------------------------------------------------------------------------
// Kernel 1: Wx[(b,t,n), o] = sum_f x[b,n,t,f] * W[f,o]   (transpose fused)
// 128 threads = 4 waves; wave = 16 rows x 64 cols, K=64 -> 64 WMMAs/wave
// ---------------------------------------------------------------------------
__global__ __launch_bounds__(128)
void k_gemm_xw(const float* __restrict__ x, const float* __restrict__ W,
               float* __restrict__ Wx) {
  __shared__ float lds[64 * LDSW];
  const int tid = threadIdx.x;
#pragma unroll
  for (int j = 0; j < 8; ++j) {                 // stage W (64x64) into LDS
    int idx = tid + j * 128;                    // float4 index 0..1023
    int row = idx >> 4, q = idx & 15;
    *(float4*)&lds[row * LDSW + q * 4] = *(const float4*)&W[row * 64 + q * 4];
  }
  __syncthreads();

  const int lane = tid & 31, wid = tid >> 5;
  const int half = lane >> 4, lmod = lane & 15;
  const int r0 = blockIdx.x * 64 + wid * 16;    // first (b,t,n) row for wave
  const int b = r0 / (Tn * Nn);
  const int t = (r0 / Nn) % Tn;
  const int n0 = r0 % Nn;
  const float* srcA = x + ((size_t)(b * Nn + n0 + lmod) * Tn + t) * Fn;

  v8f acc[4];
#pragma unroll
  for (int c = 0; c < 4; ++c)
#pragma unroll
    for (int i = 0; i < 8; ++i) acc[c][i] = 0.f;

#pragma unroll
  for (int kk = 0; kk < 16; ++kk) {
    const int k0 = kk * 4 + half * 2;           // this lane's K pair
    v2f afrag = *(const v2f*)(srcA + k0);       // A: M=lmod, K=k0,k0+1
#pragma unroll
    for (int c = 0; c < 4; ++c) {
      v2f bfrag;                                // B: N=lmod, K=k0,k0+1
      bfrag[0] = lds[k0 * LDSW + c * 16 + lmod];
      bfrag[1] = lds[(k0 + 1) * LDSW + c * 16 + lmod];
      acc[c] = wmma_f32x4(afrag, bfrag, acc[c]);
    }
  }

  float* dst = Wx + (size_t)r0 * Fn;
#pragma unroll
  for (int c = 0; c < 4; ++c)
#pragma unroll
    for (int rr = 0; rr < 8; ++rr) {
      int M = rr + half * 8;                    // C/D layout: M = vgpr + 8*half
      dst[(size_t)M * Fn + c * 16 + lmod] = acc[c][rr];
    }
}

// ---------------------------------------------------------------------------
// Kernel 2: Wx1[r] = Wx[r,:]·a[0:64], Wx2[r] = Wx[r,:]·a[64:128]
// one wave32 per row, shuffle reduction
// ---------------------------------------------------------------------------
__global__ __launch_bounds__(256)
void k_proj_a(const float* __restrict__ Wx, const float* __restrict__ avec,
              float* __restrict__ Wx1, float* __restrict__ Wx2) {
  const int lane = threadIdx.x & 31;
  const int row = blockIdx.x * 8 + (threadIdx.x >> 5);
  v2f w  = *(const v2f*)(Wx + (size_t)row * Fn + lane * 2);
  v2f a1 = *(const v2f*)(avec + lane * 2);
  v2f a2 = *(const v2f*)(avec + 64 + lane * 2);
  float p1 = w[0] * a1[0] + w[1] * a1[1];
  float p2 = w[0] * a2[0] + w[1] * a2[1];
#pragma unroll
  for (int m = 16; m >= 1; m >>= 1) {
    p1 += __shfl_xor(p1, m, 32);
    p2 += __shfl_xor(p2, m, 32);
  }
  if (lane == 0) { Wx1[row] = p1; Wx2[row] = p2; }
}

// ---------------------------------------------------------------------------
// Kernel 3: softmax statistics over the T axis (dim=1 in the reference!)
// Mx[b,n,m] = max_t s_t ; Dx[b,n,m] = sum_t exp(s_t - Mx)
// masked (adj==0) rows give s_t == NEG_BIG for all t -> att = 1/12 (faithful)
// ---------------------------------------------------------------------------
__global__ __launch_bounds__(256)
void k_softmax_t(const float* __restrict__ Wx1, const float* __restrict__ Wx2,
                 const int* __restrict__ adj, float* __restrict__ Mx,
                 float* __restrict__ Dx) {
  const int m = blockIdx.x * 256 + threadIdx.x;
  const int n = blockIdx.y;
  const int b = blockIdx.z;
  const bool masked = adj[n * Nn + m] <= 0;
  float s[Tn];
  float mx = -3.0e38f;
#pragma unroll
  for (int t = 0; t < Tn; ++t) {
    float v = Wx1[(b * Tn + t) * Nn + n] + Wx2[(b * Tn + t) * Nn + m];
    v = v > 0.f ? v : GAT_ALPHA * v;            // LeakyReLU
    if (masked) v = GAT_NEG_BIG;
    s[t] = v;
    mx = fmaxf(mx, v);
  }
  float d = 0.f;
#pragma unroll
  for (int t = 0; t < Tn; ++t) d += __expf(s[t] - mx);
  const size_t o = ((size_t)b * Nn + n) * Nn + m;
  Mx[o] = mx;
  Dx[o] = d;
}

// ---------------------------------------------------------------------------
// Kernel 4: out[b,n,t,o] = elu( sum_m att[b,t,n,m] * Wx[(b,t,m),o] )
// att recomputed on the fly into the 16x4 A fragment.
// Wx chunks double-buffered in LDS via GLOBAL_LOAD_ASYNC_TO_LDS_B128
// (ASYNCcnt pipeline: wait -> barrier -> issue k+1 -> compute k).
// ---------------------------------------------------------------------------
__global__ __launch_bounds__(128)
void k_attn_out(const float* __restrict__ Wx, const float* __restrict__ Wx1,
                const float* __restrict__ Wx2, const int* __restrict__ adj,
                const float* __restrict__ Mx, const float* __restrict__ Dx,
                float* __restrict__ out) {
  __shared__ float lds[2][64 * LDSW];           // 2 x 20 KB double buffer
  const int tid = threadIdx.x;
  const int lane = tid & 31, wid = tid >> 5;
  const int half = lane >> 4, lmod = lane & 15;
  const int r0 = blockIdx.x * 64;               // 64 n-rows of one (b,t)
  const int b = r0 / (Tn * Nn);
  const int t = (r0 / Nn) % Tn;
  const int nblk = r0 % Nn;
  const int n = nblk + wid * 16 + lmod;         // this lane's attention row

  const float  w1    = Wx1[(b * Tn + t) * Nn + n];
  const float* w2row = Wx2 + (b * Tn + t) * Nn;
  const int*   ajrow = adj + n * Nn;
  const float* mrow  = Mx + ((size_t)b * Nn + n) * Nn;
  const float* drow  = Dx + ((size_t)b * Nn + n) * Nn;
  const float* slab  = Wx + (size_t)(b * Tn + t) * Nn * Fn;

  v8f acc[4];
#pragma unroll
  for (int c = 0; c < 4; ++c)
#pragma unroll
    for (int i = 0; i < 8; ++i) acc[c][i] = 0.f;

  issue_chunk_async(slab, &lds[0][0], tid);     // prologue: chunk 0 in flight

  for (int ch = 0; ch < 8; ++ch) {
    const int mc = ch * 64;
    const int buf = ch & 1;
    asm volatile("s_wait_asynccnt 0x0" ::: "memory");  // own chunk-ch loads done
    __syncthreads();                            // chunk ch published to block;
                                                // buffer buf^1 free to overwrite
    if (ch < 7)
      issue_chunk_async(slab + (size_t)(mc + 64) * Fn, &lds[buf ^ 1][0], tid);

    const float* cur = &lds[buf][0];
#pragma unroll
    for (int ks = 0; ks < 16; ++ks) {
      const int kl = ks * 4 + half * 2;         // local K pair in chunk
      const int m = mc + kl;
      v2f w2 = *(const v2f*)(w2row + m);
      v2f mv = *(const v2f*)(mrow + m);
      v2f dv = *(const v2f*)(drow + m);
      const int aj0 = ajrow[m], aj1 = ajrow[m + 1];
      float s0 = w1 + w2[0]; s0 = s0 > 0.f ? s0 : GAT_ALPHA * s0;
      float s1 = w1 + w2[1]; s1 = s1 > 0.f ? s1 : GAT_ALPHA * s1;
      if (aj0 <= 0) s0 = GAT_NEG_BIG;
      if (aj1 <= 0) s1 = GAT_NEG_BIG;
      v2f afrag;                                // att A-fragment (16x4)
      afrag[0] = __fdividef(__expf(s0 - mv[0]), dv[0]);
      afrag[1] = __fdividef(__expf(s1 - mv[1]), dv[1]);
#pragma unroll
      for (int c = 0; c < 4; ++c) {
        v2f bfrag;
        bfrag[0] = cur[kl * LDSW + c * 16 + lmod];
        bfrag[1] = cur[(kl + 1) * LDSW + c * 16 + lmod];
        acc[c] = wmma_f32x4(afrag, bfrag, acc[c]);
      }
    }
  }

  // ELU + store with (B,T,N,F) -> (B,N,T,F) transpose fused
#pragma unroll
  for (int c = 0; c < 4; ++c)
#pragma unroll
    for (int rr = 0; rr < 8; ++rr) {
      int M = rr + half * 8;
      int nout = nblk + wid * 16 + M;
      float v = acc[c][rr];
      v = v > 0.f ? v : (__expf(v) - 1.f);      // elu, alpha = 1
      out[((size_t)(b * Nn + nout) * Tn + t) * Fn + c * 16 + lmod] = v;
    }
}

// ---------------------------------------------------------------------------
extern "C" void kernel_launch(void* const* d_in, const int* in_sizes, int n_in,
                              void* d_out, int out_size, void* d_ws,
                              size_t ws_size, hipStream_t stream) {
  const float* x   = (const float*)d_in[0];   // (B, N, T, F)
  const int*   adj = (const int*)  d_in[1];   // (N, N)
  const float* W   = (const float*)d_in[2];   // (F, F)
  const float* a   = (const float*)d_in[3];   // (2F, 1)
  float* out = (float*)d_out;                 // (B, N, T, F)
  float* ws  = (float*)d_ws;

  float* Wx  = ws;                            // R * F floats (50.3 MB)
  float* Wx1 = Wx + (size_t)Rn * Fn;          // R floats
  float* Wx2 = Wx1 + Rn;                      // R floats
  float* Mx  = Wx2 + Rn;                      // B*N*N floats (33.5 MB)
  float* Dx  = Mx + (size_t)Bn * Nn * Nn;     // B*N*N floats (33.5 MB)

  hipLaunchKernelGGL(k_gemm_xw,  dim3(Rn / 64), dim3(128), 0, stream, x, W, Wx);
  hipLaunchKernelGGL(k_proj_a,   dim3(Rn / 8),  dim3(256), 0, stream, Wx, a, Wx1, Wx2);
  hipLaunchKernelGGL(k_softmax_t, dim3(Nn / 256, Nn, Bn), dim3(256), 0, stream,
                     Wx1, Wx2, adj, Mx, Dx);
  hipLaunchKernelGGL(k_attn_out, dim3(Rn / 64), dim3(128), 0, stream,
                     Wx, Wx1, Wx2, adj, Mx, Dx, out);
}